// DecomposedRelativePositionEncoder_38268158607457
// MI455X (gfx1250) — compile-verified
//
#include <hip/hip_runtime.h>

typedef __attribute__((ext_vector_type(2))) float v2f;
typedef __attribute__((ext_vector_type(4))) float v4f;
typedef __attribute__((ext_vector_type(8))) float v8f;

#define CC 80          // channels
#define RSTR 68        // LDS row stride in floats (68*4 = 272 bytes, 16B aligned)

// One workgroup per (b, hh): computes the 64-row bias tiles with f32 WMMA,
// then streams the (64 x 4096) broadcast-add block with NT stores.
__global__ __launch_bounds__(256) void relpos_fused_kernel(
    const float* __restrict__ q,     // (16, 4096, 80)
    const float* __restrict__ relH,  // (127, 80)
    const float* __restrict__ relW,  // (127, 80)
    float* __restrict__ out)         // (16, 4096, 4096)
{
    __shared__ float rh_l[64 * RSTR];   // rh[w][j]
    __shared__ float rw_l[64 * RSTR];   // rw[w][i]

    const int tid  = threadIdx.x;
    const int lane = tid & 31;
    const int wv   = tid >> 5;       // wave id 0..7 (wave32)
    const int half = lane >> 4;      // 0 or 1
    const int lm   = lane & 15;

    const int b  = blockIdx.x >> 6;
    const int hh = blockIdx.x & 63;

    // warm the tiny relpos tables into cache (gfx1250 global_prefetch_b8)
    if (tid * 40 < 127 * CC) {
        __builtin_prefetch(relH + tid * 40, 0, 3);
        __builtin_prefetch(relW + tid * 40, 0, 3);
    }

    const float* qBase = q + (size_t)((b * 64 + hh) * 64) * CC;

    // 36 WMMA tiles per workgroup:
    //   t in [0,16):  Gh tile, wt = t>>2, dt = t&3,   d = hh + 16*dt + n   (always in range)
    //   t in [16,36): Gw tile, wt = u/5,  dt = u%5,   d = 16*wt + 16*dt + n (clamped)
    for (int t = wv; t < 36; t += 8) {
        int wt, dt, dRow;
        const float* relBase;
        bool isH;
        if (t < 16) {
            wt = t >> 2; dt = t & 3;
            dRow = hh + dt * 16 + lm;            // <= 126 always
            relBase = relH;
            isH = true;
        } else {
            int u = t - 16;
            wt = u / 5; dt = u % 5;
            dRow = wt * 16 + dt * 16 + lm;
            if (dRow > 126) dRow = 126;          // clamp; remap discards it below
            relBase = relW;
            isH = false;
        }
        const int wBase = wt * 16;

        // A fragment: q row (wBase+lm), cols k0 + 2*half + {0,1}
        // B fragment: table row dRow,   cols k0 + 2*half + {0,1}
        const float* aPtr = qBase   + (size_t)(wBase + lm) * CC + 2 * half;
        const float* bPtr = relBase + (size_t)dRow        * CC + 2 * half;

        v8f acc = {0.f, 0.f, 0.f, 0.f, 0.f, 0.f, 0.f, 0.f};
        #pragma unroll
        for (int k0 = 0; k0 < CC; k0 += 4) {
            v2f a  = *(const v2f*)(aPtr + k0);
            v2f bb = *(const v2f*)(bPtr + k0);
            // (neg_a, A, neg_b, B, c_mod, C, reuse_a, reuse_b)
            acc = __builtin_amdgcn_wmma_f32_16x16x4_f32(
                false, a, false, bb, (short)0, acc, false, false);
        }

        // D layout: lane holds N = lm, rows M = v + 8*half
        if (isH) {
            const int j = 63 - dt * 16 - lm;     // j = hh+63-d, always in [0,63]
            #pragma unroll
            for (int v = 0; v < 8; ++v) {
                int wrow = wBase + v + 8 * half;
                rh_l[wrow * RSTR + j] = acc[v];
            }
        } else {
            const int dcol = wt * 16 + dt * 16 + lm;
            #pragma unroll
            for (int v = 0; v < 8; ++v) {
                int wrow = wBase + v + 8 * half;
                int i = wrow + 63 - dcol;        // i = w+63-d
                if (i >= 0 && i < 64)
                    rw_l[wrow * RSTR + i] = acc[v];
            }
        }
    }

    __syncthreads();

    // Phase 2: stream out[b, hh*64 + w, j*64 + i] = rh[w][j] + rw[w][i]
    // 1 MiB per workgroup; coalesced 16B NT stores, LDS reads are broadcasts.
    float* outBase = out + (size_t)(b * 4096 + hh * 64) * 4096;
    #pragma unroll 1
    for (int w = 0; w < 64; ++w) {
        const float* rhRow = &rh_l[w * RSTR];
        const float* rwRow = &rw_l[w * RSTR];
        float* orow = outBase + (size_t)w * 4096;
        #pragma unroll
        for (int p = 0; p < 4; ++p) {
            int col   = p * 1024 + tid * 4;      // 4 KiB contiguous per pass
            float rhv = rhRow[col >> 6];         // j broadcast (uniform per 16 lanes)
            v4f  rwv  = *(const v4f*)(rwRow + (col & 63)); // 16B-aligned, same-addr bcast
            v4f  o;
            o.x = rhv + rwv.x;
            o.y = rhv + rwv.y;
            o.z = rhv + rwv.z;
            o.w = rhv + rwv.w;
            __builtin_nontemporal_store(o, (v4f*)(orow + col));
        }
    }
}

extern "C" void kernel_launch(void* const* d_in, const int* in_sizes, int n_in,
                              void* d_out, int out_size, void* d_ws, size_t ws_size,
                              hipStream_t stream) {
    const float* q    = (const float*)d_in[0];   // (16, 4096, 80)
    const float* relH = (const float*)d_in[1];   // (127, 80)
    const float* relW = (const float*)d_in[2];   // (127, 80)
    // d_in[3], d_in[4] are h, w scalars; shapes are fixed at 64 in this harness.
    float* out = (float*)d_out;                  // (16, 4096, 4096)

    dim3 grid(16 * 64);   // one workgroup per (batch, grid-row)
    dim3 block(256);      // 8 wave32s
    hipLaunchKernelGGL(relpos_fused_kernel, grid, block, 0, stream,
                       q, relH, relW, out);
}